// S4_39616778338840
// MI455X (gfx1250) — compile-verified
//
#include <hip/hip_runtime.h>
#include <math.h>

// Problem dims
#define B_SZ   32
#define T_LEN  2048
#define DIN    64
#define P_DIM  256
#define H_DIM  512
#define DOUT   64
#define M_ROWS (B_SZ * T_LEN)   // 65536 rows of (b,t)

// Scan chunking
#define NC 32                   // number of chunks
#define CL 64                   // chunk length = T_LEN / NC  (a^CL via 6 squarings)

typedef float v2f __attribute__((ext_vector_type(2)));
typedef float v8f __attribute__((ext_vector_type(8)));
typedef unsigned int u32x4 __attribute__((ext_vector_type(4)));
typedef int i32x4 __attribute__((ext_vector_type(4)));
typedef int i32x8 __attribute__((ext_vector_type(8)));

__device__ __forceinline__ v8f wmma_f32_k4(v2f a, v2f b, v8f c) {
  // D = A(16x4 f32) x B(4x16 f32) + C(16x16 f32)
  return __builtin_amdgcn_wmma_f32_16x16x4_f32(
      /*neg_a=*/false, a, /*neg_b=*/false, b,
      /*c_mod=*/(short)0, c, /*reuse_a=*/false, /*reuse_b=*/false);
}

__device__ __forceinline__ float gelu_tanh(float x) {
  // jax.nn.gelu (approximate=True): 0.5*x*(1+tanh(y)) == x * sigmoid(2y)
  // Branchless: avoids the divergent EXEC save/restore of libm tanhf.
  const float c0 = 0.7978845608028654f;   // sqrt(2/pi)
  const float c1 = 0.044715f;
  float y = c0 * (x + c1 * x * x * x);
  return __fdividef(x, 1.0f + __expf(-2.0f * y));
}

// ---------------------------------------------------------------------------
// CW = C (H x P) @ W_out^T (P x DOUT)  -> (H x DOUT). Tiny: scalar dot per elt.
// ---------------------------------------------------------------------------
__global__ void k_cw(const float* __restrict__ C, const float* __restrict__ Wout,
                     float* __restrict__ CW) {
  int idx = blockIdx.x * blockDim.x + threadIdx.x;  // H*DOUT = 32768
  int o = idx & (DOUT - 1);
  int h = idx >> 6;
  const float* cr = C + (size_t)h * P_DIM;
  const float* wr = Wout + (size_t)o * P_DIM;
  float s = 0.0f;
  for (int p = 0; p < P_DIM; ++p) s = fmaf(cr[p], wr[p], s);
  CW[(size_t)h * DOUT + o] = s;
}

// ---------------------------------------------------------------------------
// U[m, :] = gelu(x[m,:] @ W_in^T + b_in) @ Bmat      (fused, 16-row tiles)
// One wave per 16-row tile; gelu'd p-tile staged in LDS between the two GEMMs.
// ---------------------------------------------------------------------------
#define LDS_STRIDE 260   // 256 + pad: bank-conflict-free column reads
__global__ void k_in(const float* __restrict__ x, const float* __restrict__ Win,
                     const float* __restrict__ bin, const float* __restrict__ Bmat,
                     float* __restrict__ U) {
  __shared__ float ldsP[2 * 16 * LDS_STRIDE];   // 2 waves/block, 16x256 each
  const int lane = threadIdx.x & 31;
  const int wave = threadIdx.x >> 5;
  const int lm   = lane & 15;          // row (A) / col (B,D) within 16
  const int half = lane >> 4;          // K-pair select for A/B frags
  const int kk   = 2 * half;
  const int m0   = (blockIdx.x * 2 + wave) * 16;
  float* myP = ldsP + wave * 16 * LDS_STRIDE;

  // ---- Stage 1: p = gelu(x @ Win^T + b_in), tile 16 x 256 ----
  for (int nt = 0; nt < P_DIM / 16; ++nt) {
    int n0 = nt * 16;
    float bias = bin[n0 + lm];
    v8f acc;
#pragma unroll
    for (int i = 0; i < 8; ++i) acc[i] = bias;
#pragma unroll
    for (int k0 = 0; k0 < DIN; k0 += 4) {
      v2f a, b;
      const float* ap = x + (size_t)(m0 + lm) * DIN + k0 + kk;
      a.x = ap[0]; a.y = ap[1];
      // B-frag element (k, n) = Win[n][k]
      const float* bp = Win + (size_t)(n0 + lm) * DIN + k0 + kk;
      b.x = bp[0]; b.y = bp[1];
      acc = wmma_f32_k4(a, b, acc);
    }
#pragma unroll
    for (int r = 0; r < 8; ++r)
      myP[(r + half * 8) * LDS_STRIDE + n0 + lm] = gelu_tanh(acc[r]);
  }
  __syncthreads();

  // ---- Stage 2: U-tile = p @ Bmat, tile 16 x 512 ----
  for (int nt = 0; nt < H_DIM / 16; ++nt) {
    int n0 = nt * 16;
    v8f acc;
#pragma unroll
    for (int i = 0; i < 8; ++i) acc[i] = 0.0f;
    for (int k0 = 0; k0 < P_DIM; k0 += 4) {
      v2f a, b;
      const float* ap = myP + lm * LDS_STRIDE + k0 + kk;
      a.x = ap[0]; a.y = ap[1];
      b.x = Bmat[(size_t)(k0 + kk) * H_DIM + n0 + lm];
      b.y = Bmat[(size_t)(k0 + kk + 1) * H_DIM + n0 + lm];
      acc = wmma_f32_k4(a, b, acc);
    }
#pragma unroll
    for (int r = 0; r < 8; ++r)
      U[(size_t)(m0 + r + half * 8) * H_DIM + n0 + lm] = acc[r];
  }
}

// ---------------------------------------------------------------------------
// Scan pass 1: per (b, chunk, h), local recurrence from 0; write chunk total.
// ---------------------------------------------------------------------------
__global__ void k_scan1(const float* __restrict__ U, const float* __restrict__ a,
                        float* __restrict__ totals) {
  int idx = blockIdx.x * blockDim.x + threadIdx.x;   // B*NC*H = 524288
  int h = idx & (H_DIM - 1);
  int c = (idx >> 9) & (NC - 1);
  int b = idx >> 14;
  float ah = a[h];
  const float* up = U + ((size_t)(b * T_LEN + c * CL)) * H_DIM + h;
  float acc = 0.0f;
#pragma unroll 4
  for (int t = 0; t < CL; ++t) {
    if (t + 16 < CL) __builtin_prefetch(up + (size_t)(t + 16) * H_DIM, 0, 1);
    acc = fmaf(ah, acc, up[(size_t)t * H_DIM]);
  }
  totals[((size_t)b * NC + c) * H_DIM + h] = acc;
}

// ---------------------------------------------------------------------------
// Scan pass 2: per (b, h), combine chunk totals into per-chunk carry-ins.
// ---------------------------------------------------------------------------
__global__ void k_scan2(const float* __restrict__ totals, const float* __restrict__ a,
                        float* __restrict__ carries) {
  int idx = blockIdx.x * blockDim.x + threadIdx.x;   // B*H = 16384
  int h = idx & (H_DIM - 1);
  int b = idx >> 9;
  float ah = a[h];
  float aL = ah;
#pragma unroll
  for (int i = 0; i < 6; ++i) aL = aL * aL;          // a^64 == a^CL
  float carry = 0.0f;
  for (int c = 0; c < NC; ++c) {
    size_t off = ((size_t)b * NC + c) * H_DIM + h;
    carries[off] = carry;
    carry = fmaf(aL, carry, totals[off]);
  }
}

// ---------------------------------------------------------------------------
// Scan pass 3: re-run each chunk seeded with its carry; write h in place of U.
// ---------------------------------------------------------------------------
__global__ void k_scan3(float* __restrict__ U, const float* __restrict__ a,
                        const float* __restrict__ carries) {
  int idx = blockIdx.x * blockDim.x + threadIdx.x;   // B*NC*H
  int h = idx & (H_DIM - 1);
  int c = (idx >> 9) & (NC - 1);
  int b = idx >> 14;
  float ah = a[h];
  float* up = U + ((size_t)(b * T_LEN + c * CL)) * H_DIM + h;
  float acc = carries[((size_t)b * NC + c) * H_DIM + h];
#pragma unroll 4
  for (int t = 0; t < CL; ++t) {
    if (t + 16 < CL) __builtin_prefetch(up + (size_t)(t + 16) * H_DIM, 0, 1);
    acc = fmaf(ah, acc, up[(size_t)t * H_DIM]);
    up[(size_t)t * H_DIM] = acc;
  }
}

// ---------------------------------------------------------------------------
// out = h @ CW + b_out   (M x 512) x (512 x 64).
// CW staged in LDS in 32KB K-chunks via the Tensor Data Mover (TDM):
// wave 0 issues TENSOR_LOAD_TO_LDS with a hand-built D#, waits TENSORcnt,
// then the workgroup barriers. One wave per 16-row tile, 4 waves/block.
// ---------------------------------------------------------------------------
__global__ void k_out(const float* __restrict__ Hbuf, const float* __restrict__ CW,
                      const float* __restrict__ bout, float* __restrict__ out) {
  __shared__ float ldsW[128 * DOUT];                 // 32 KB chunk of CW (LDS offset 0)
  const int lane = threadIdx.x & 31;
  const int wave = threadIdx.x >> 5;
  const int lm   = lane & 15;
  const int half = lane >> 4;
  const int kk   = 2 * half;
  const int m0   = (blockIdx.x * 4 + wave) * 16;

  v8f acc0, acc1, acc2, acc3;
#pragma unroll
  for (int i = 0; i < 8; ++i) { acc0[i] = 0.f; acc1[i] = 0.f; acc2[i] = 0.f; acc3[i] = 0.f; }

  for (int kc = 0; kc < H_DIM / 128; ++kc) {
    if (wave == 0) {
      // ---- D# descriptor: 1-D tile of 8192 dwords, global -> LDS offset 0 ----
      unsigned long long ga =
          (unsigned long long)(const void*)(CW + (size_t)kc * 128 * DOUT);
      u32x4 g0;
      g0[0] = 1u;                                   // count=1 (valid descriptor)
      g0[1] = 0u;                                   // lds_addr = 0
      g0[2] = (unsigned)(ga & 0xffffffffu);         // global_addr[31:0]
      g0[3] = (unsigned)((ga >> 32) & 0x01ffffffu)  // global_addr[56:32]
              | (2u << 30);                         // type = 2 ("image")
      i32x8 g1;
      g1[0] = (int)(2u << 16);                      // data_size = 4 bytes
      g1[1] = (int)((8192u & 0xffffu) << 16);       // tensor_dim0[15:0]
      g1[2] = (int)((8192u >> 16) | (1u << 16));    // tensor_dim0[31:16] | tensor_dim1=1
      g1[3] = (int)(8192u << 16);                   // tile_dim0 = 8192 dwords
      g1[4] = 1;                                    // tile_dim1 = 1, tile_dim2 = 0
      g1[5] = 8192;                                 // tensor_dim0_stride[31:0]
      g1[6] = 0;
      g1[7] = 0;
      i32x4 gz4a; gz4a[0] = 0; gz4a[1] = 0; gz4a[2] = 0; gz4a[3] = 0;
      i32x4 gz4b; gz4b[0] = 0; gz4b[1] = 0; gz4b[2] = 0; gz4b[3] = 0;
      i32x8 gz8;
#pragma unroll
      for (int i = 0; i < 8; ++i) gz8[i] = 0;
      // 6-arg toolchain variant: (g0, g1, g2, g3, g_extra, cpol)
      __builtin_amdgcn_tensor_load_to_lds(g0, g1, gz4a, gz4b, gz8, 0);
      __builtin_amdgcn_s_wait_tensorcnt(0);
    }
    __syncthreads();

    for (int k0 = 0; k0 < 128; k0 += 4) {
      v2f a;
      const float* ap = Hbuf + (size_t)(m0 + lm) * H_DIM + kc * 128 + k0 + kk;
      a.x = ap[0]; a.y = ap[1];
      const float* r0 = ldsW + (k0 + kk) * DOUT;
      const float* r1 = ldsW + (k0 + kk + 1) * DOUT;
      v2f b0, b1, b2, b3;
      b0.x = r0[lm];      b0.y = r1[lm];
      b1.x = r0[16 + lm]; b1.y = r1[16 + lm];
      b2.x = r0[32 + lm]; b2.y = r1[32 + lm];
      b3.x = r0[48 + lm]; b3.y = r1[48 + lm];
      acc0 = wmma_f32_k4(a, b0, acc0);
      acc1 = wmma_f32_k4(a, b1, acc1);
      acc2 = wmma_f32_k4(a, b2, acc2);
      acc3 = wmma_f32_k4(a, b3, acc3);
    }
    __syncthreads();
  }

  float bb0 = bout[lm], bb1 = bout[16 + lm], bb2 = bout[32 + lm], bb3 = bout[48 + lm];
#pragma unroll
  for (int r = 0; r < 8; ++r) {
    size_t row = (size_t)(m0 + r + half * 8) * DOUT;
    out[row + lm]       = acc0[r] + bb0;
    out[row + 16 + lm]  = acc1[r] + bb1;
    out[row + 32 + lm]  = acc2[r] + bb2;
    out[row + 48 + lm]  = acc3[r] + bb3;
  }
}

// ---------------------------------------------------------------------------
extern "C" void kernel_launch(void* const* d_in, const int* in_sizes, int n_in,
                              void* d_out, int out_size, void* d_ws, size_t ws_size,
                              hipStream_t stream) {
  const float* x    = (const float*)d_in[0];  // [B,T,DIN]
  const float* a    = (const float*)d_in[1];  // [H]
  const float* Bm   = (const float*)d_in[2];  // [P,H]
  const float* Cm   = (const float*)d_in[3];  // [H,P]
  const float* Win  = (const float*)d_in[4];  // [P,DIN]
  const float* bin  = (const float*)d_in[5];  // [P]
  const float* Wout = (const float*)d_in[6];  // [DOUT,P]
  const float* bout = (const float*)d_in[7];  // [DOUT]
  float* out = (float*)d_out;                 // [B,T,DOUT]

  char* ws = (char*)d_ws;
  const size_t U_BYTES   = (size_t)M_ROWS * H_DIM * sizeof(float);      // 128 MB
  const size_t TOT_BYTES = (size_t)B_SZ * NC * H_DIM * sizeof(float);   // 2 MB
  float* U       = (float*)(ws);
  float* totals  = (float*)(ws + U_BYTES);
  float* carries = (float*)(ws + U_BYTES + TOT_BYTES);
  float* CW      = (float*)(ws + U_BYTES + 2 * TOT_BYTES);              // 128 KB

  // 1) fold output projections: CW = C @ W_out^T
  k_cw<<<dim3((H_DIM * DOUT) / 256), dim3(256), 0, stream>>>(Cm, Wout, CW);
  // 2) U = gelu(x @ Win^T + b_in) @ B   (WMMA, fused)
  k_in<<<dim3(M_ROWS / 32), dim3(64), 0, stream>>>(x, Win, bin, Bm, U);
  // 3) chunked diagonal scan (3 passes), h overwrites U
  k_scan1<<<dim3((B_SZ * NC * H_DIM) / 256), dim3(256), 0, stream>>>(U, a, totals);
  k_scan2<<<dim3((B_SZ * H_DIM) / 256), dim3(256), 0, stream>>>(totals, a, carries);
  k_scan3<<<dim3((B_SZ * NC * H_DIM) / 256), dim3(256), 0, stream>>>(U, a, carries);
  // 4) out = h @ CW + b_out   (WMMA + TDM staging)
  k_out<<<dim3(M_ROWS / 64), dim3(128), 0, stream>>>(U, CW, bout, out);
}